// MultiHeadSelfAttention_52570399703362
// MI455X (gfx1250) — compile-verified
//
#include <hip/hip_runtime.h>
#include <math.h>

#define BB 2
#define SS 2048
#define EE 1024
#define HH 16
#define DD 64
#define KS 4   // split-K factor for the two streaming WMMA kernels

typedef __attribute__((ext_vector_type(2))) float v2f;
typedef __attribute__((ext_vector_type(8))) float v8f;

// ---------------------------------------------------------------------------
// K1: q[b][e'] = sum_e inputs[b, S-1, e] * Wq[e', e]
// ---------------------------------------------------------------------------
__global__ void qproj_kernel(const float* __restrict__ inputs,
                             const float* __restrict__ Wq,
                             float* __restrict__ q) {
    __shared__ float xs[EE];
    int gid = blockIdx.x * 256 + threadIdx.x;
    int b   = gid >> 10;          // EE == 1024
    int ep  = gid & (EE - 1);
    const float* xlast = inputs + ((size_t)b * SS + (SS - 1)) * EE;
    for (int i = threadIdx.x; i < EE; i += 256) xs[i] = xlast[i];
    __syncthreads();
    const float* wrow = Wq + (size_t)ep * EE;
    float acc = 0.f;
    for (int k = 0; k < EE; k += 4) {
        float4 w = *(const float4*)(wrow + k);
        acc = fmaf(w.x, xs[k],     acc);
        acc = fmaf(w.y, xs[k + 1], acc);
        acc = fmaf(w.z, xs[k + 2], acc);
        acc = fmaf(w.w, xs[k + 3], acc);
    }
    q[gid] = acc;
}

// ---------------------------------------------------------------------------
// K2: rT[b][e][h] = sum_d Wk[h*64+d, e] * q[b][h*64+d]   (stored e-major, h minor)
// ---------------------------------------------------------------------------
__global__ void rproj_kernel(const float* __restrict__ Wk,
                             const float* __restrict__ q,
                             float* __restrict__ rT) {
    __shared__ float qs[DD];
    int gid = blockIdx.x * 256 + threadIdx.x;
    int b   = gid >> 14;          // HH*EE == 16384
    int rem = gid & 16383;
    int h   = rem >> 10;
    int e   = rem & 1023;
    if (threadIdx.x < DD) qs[threadIdx.x] = q[b * EE + h * DD + threadIdx.x];
    __syncthreads();
    float acc = 0.f;
    for (int d = 0; d < DD; ++d)
        acc = fmaf(Wk[((size_t)(h * DD + d)) * EE + e], qs[d], acc);
    rT[((size_t)b * EE + e) * HH + h] = acc;
}

// ---------------------------------------------------------------------------
// K3: scores (split-K): scp[ks][b][h][s] = partial  embK[b,s,:] . r[b,h,:]
// GEMM per batch: [S x E] @ [E x H], M-tile=16, N=16 (one WMMA tile), f32 WMMA
// grid: (b * 128 + mt) * KS + ks, 32 threads (1 wave)
// ---------------------------------------------------------------------------
__global__ void __launch_bounds__(32)
scores_kernel(const float* __restrict__ embK,
              const float* __restrict__ rT,
              float* __restrict__ scp) {
    int blk = blockIdx.x;
    int ks  = blk & (KS - 1);
    int t   = blk >> 2;           // KS == 4
    int b   = t >> 7;             // 128 m-tiles per batch
    int mt  = t & 127;
    int m0  = mt * 16;
    int lane = threadIdx.x;
    int lh = lane >> 4;           // 0: lanes 0-15, 1: lanes 16-31
    int lm = lane & 15;

    // A fragment base: 16x4 f32, lane=M, float2 = K{0,1} (+2 for upper half)
    const float* A  = embK + ((size_t)b * SS + m0 + lm) * EE + 2 * lh;
    // B fragment base: 4x16 f32, lane (mod 16)=N, VGPR j -> K = j + 2*lh
    const float* Bp = rT + (size_t)b * EE * HH + 32 * lh + lm;

    v8f c = {0.f, 0.f, 0.f, 0.f, 0.f, 0.f, 0.f, 0.f};
    int kbeg = ks * (EE / 4 / KS);          // 64 k-steps per split
    int kend = kbeg + (EE / 4 / KS);
    for (int kk = kbeg; kk < kend; ++kk) {
        v2f a = *(const v2f*)(A + 4 * kk);
        v2f bb;
        bb[0] = Bp[64 * kk];
        bb[1] = Bp[64 * kk + 16];
        c = __builtin_amdgcn_wmma_f32_16x16x4_f32(false, a, false, bb,
                                                  (short)0, c, false, false);
    }
    // D layout: lane l, vgpr j -> M = j + 8*lh, N = lm (head index)
    float* dst = scp + (size_t)ks * BB * HH * SS;
    for (int j = 0; j < 8; ++j) {
        int m = j + 8 * lh;
        dst[((size_t)(b * HH + lm)) * SS + m0 + m] = c[j];
    }
}

// ---------------------------------------------------------------------------
// K4: softmax over S per (b,h); sums KS partials first (fixed order = determ.)
// ---------------------------------------------------------------------------
__global__ void softmax_kernel(const float* __restrict__ scp,
                               float* __restrict__ p) {
    __shared__ float red[256];
    int row = blockIdx.x;                    // b*HH + h
    size_t base = (size_t)row * SS;
    const size_t cs = (size_t)BB * HH * SS;  // chunk stride
    int t = threadIdx.x;
    float v[8];
    float vmax = -3.4e38f;
    for (int i = 0; i < 8; ++i) {
        size_t off = base + t + i * 256;
        float s = scp[off] + scp[off + cs] + scp[off + 2 * cs] + scp[off + 3 * cs];
        v[i] = s;
        vmax = fmaxf(vmax, s);
    }
    red[t] = vmax; __syncthreads();
    for (int o = 128; o > 0; o >>= 1) {
        if (t < o) red[t] = fmaxf(red[t], red[t + o]);
        __syncthreads();
    }
    vmax = red[0]; __syncthreads();
    float sum = 0.f;
    for (int i = 0; i < 8; ++i) { v[i] = expf(v[i] - vmax); sum += v[i]; }
    red[t] = sum; __syncthreads();
    for (int o = 128; o > 0; o >>= 1) {
        if (t < o) red[t] += red[t + o];
        __syncthreads();
    }
    float inv = 1.f / red[0];
    for (int i = 0; i < 8; ++i) p[base + t + i * 256] = v[i] * inv;
}

// ---------------------------------------------------------------------------
// K5: u (split-K): up[ks][b][h][e] = partial  sum_s p[b,h,s] * embV[b,s,e]
// GEMM per batch: [H x S] @ [S x E], M=16 (one tile), N-tile=16
// grid: (b * 64 + nt) * KS + ks, 32 threads
// ---------------------------------------------------------------------------
__global__ void __launch_bounds__(32)
uproj_kernel(const float* __restrict__ embV,
             const float* __restrict__ p,
             float* __restrict__ up) {
    int blk = blockIdx.x;
    int ks  = blk & (KS - 1);
    int t   = blk >> 2;
    int b   = t >> 6;             // 64 n-tiles per batch
    int nt  = t & 63;
    int n0  = nt * 16;
    int lane = threadIdx.x;
    int lh = lane >> 4;
    int lm = lane & 15;

    // A fragment: p is [H x S], lane=M(head), float2 = K{0,1}(+2 upper half)
    const float* A  = p + ((size_t)b * HH + lm) * SS + 2 * lh;
    // B fragment: embV rows are K, cols are N
    const float* Bp = embV + (size_t)b * SS * EE + (size_t)(2 * lh) * EE + n0 + lm;

    v8f c = {0.f, 0.f, 0.f, 0.f, 0.f, 0.f, 0.f, 0.f};
    int kbeg = ks * (SS / 4 / KS);          // 128 k-steps per split
    int kend = kbeg + (SS / 4 / KS);
    for (int kk = kbeg; kk < kend; ++kk) {
        v2f a = *(const v2f*)(A + 4 * kk);
        v2f bb;
        bb[0] = Bp[(size_t)4 * kk * EE];
        bb[1] = Bp[(size_t)4 * kk * EE + EE];
        c = __builtin_amdgcn_wmma_f32_16x16x4_f32(false, a, false, bb,
                                                  (short)0, c, false, false);
    }
    float* dst = up + (size_t)ks * BB * HH * EE;
    for (int j = 0; j < 8; ++j) {
        int m = j + 8 * lh;                 // head index
        dst[((size_t)(b * HH + m)) * EE + n0 + lm] = c[j];
    }
}

// ---------------------------------------------------------------------------
// K6: o[b][h*64+d] = sum_e Wv[h*64+d, e] * u[b][h][e]  (sums KS partials)
// ---------------------------------------------------------------------------
__global__ void oproj_kernel(const float* __restrict__ Wv,
                             const float* __restrict__ up,
                             float* __restrict__ o) {
    __shared__ float us[4 * EE];           // 4 heads per block, 16 KB
    int gid = blockIdx.x * 256 + threadIdx.x;
    int b   = gid >> 10;
    int idx = gid & 1023;
    int h   = idx >> 6;
    int d   = idx & 63;
    int hbase = (blockIdx.x & 3) * 4;      // first head handled by this block
    const size_t cs = (size_t)BB * HH * EE;
    for (int i = threadIdx.x; i < 4 * EE; i += 256) {
        size_t off = ((size_t)b * HH + hbase + (i >> 10)) * EE + (i & 1023);
        us[i] = up[off] + up[off + cs] + up[off + 2 * cs] + up[off + 3 * cs];
    }
    __syncthreads();
    const float* wrow = Wv + (size_t)(h * DD + d) * EE;
    const float* urow = us + (h - hbase) * EE;
    float acc = 0.f;
    for (int k = 0; k < EE; k += 4) {
        float4 w = *(const float4*)(wrow + k);
        acc = fmaf(w.x, urow[k],     acc);
        acc = fmaf(w.y, urow[k + 1], acc);
        acc = fmaf(w.z, urow[k + 2], acc);
        acc = fmaf(w.w, urow[k + 3], acc);
    }
    o[gid] = acc;
}

// ---------------------------------------------------------------------------
// K7: orow[b][e'] = sum_e Wo[e', e] * o[b][e]
// ---------------------------------------------------------------------------
__global__ void outrow_kernel(const float* __restrict__ Wo,
                              const float* __restrict__ o,
                              float* __restrict__ orow) {
    __shared__ float os[EE];
    int gid = blockIdx.x * 256 + threadIdx.x;
    int b   = gid >> 10;
    int ep  = gid & 1023;
    for (int i = threadIdx.x; i < EE; i += 256) os[i] = o[b * EE + i];
    __syncthreads();
    const float* wrow = Wo + (size_t)ep * EE;
    float acc = 0.f;
    for (int k = 0; k < EE; k += 4) {
        float4 w = *(const float4*)(wrow + k);
        acc = fmaf(w.x, os[k],     acc);
        acc = fmaf(w.y, os[k + 1], acc);
        acc = fmaf(w.z, os[k + 2], acc);
        acc = fmaf(w.w, os[k + 3], acc);
    }
    orow[gid] = acc;
}

// ---------------------------------------------------------------------------
// K8: broadcast orow to all S positions: out[b][s][:] = orow[b][:]
// ---------------------------------------------------------------------------
__global__ void bcast_kernel(const float* __restrict__ orow,
                             float4* __restrict__ out) {
    size_t gid = (size_t)blockIdx.x * 256 + threadIdx.x;   // B*S*E/4 total
    const int perB = SS * (EE / 4);
    int b   = (int)(gid / perB);
    int e4  = (int)(gid & (EE / 4 - 1));
    const float4* r4 = (const float4*)orow;
    out[gid] = r4[b * (EE / 4) + e4];
}

// ---------------------------------------------------------------------------
extern "C" void kernel_launch(void* const* d_in, const int* in_sizes, int n_in,
                              void* d_out, int out_size, void* d_ws, size_t ws_size,
                              hipStream_t stream) {
    const float* inputs = (const float*)d_in[0];
    const float* embV   = (const float*)d_in[1];
    const float* embK   = (const float*)d_in[2];
    const float* Wq     = (const float*)d_in[3];
    const float* Wk     = (const float*)d_in[4];
    const float* Wv     = (const float*)d_in[5];
    const float* Wo     = (const float*)d_in[6];

    float* ws  = (float*)d_ws;
    float* q    = ws;                                  // B*E            = 2048
    float* rT   = q    + (size_t)BB * EE;              // B*E*H          = 32768
    float* scp  = rT   + (size_t)BB * EE * HH;         // KS*B*H*S       = 262144
    float* p    = scp  + (size_t)KS * BB * HH * SS;    // B*H*S          = 65536
    float* up   = p    + (size_t)BB * HH * SS;         // KS*B*H*E       = 131072
    float* o    = up   + (size_t)KS * BB * HH * EE;    // B*E            = 2048
    float* orow = o    + (size_t)BB * EE;              // B*E            = 2048

    qproj_kernel  <<<(BB * EE) / 256, 256, 0, stream>>>(inputs, Wq, q);
    rproj_kernel  <<<(BB * HH * EE) / 256, 256, 0, stream>>>(Wk, q, rT);
    scores_kernel <<<BB * 128 * KS, 32, 0, stream>>>(embK, rT, scp);
    softmax_kernel<<<BB * HH, 256, 0, stream>>>(scp, p);
    uproj_kernel  <<<BB * 64 * KS, 32, 0, stream>>>(embV, p, up);
    oproj_kernel  <<<(BB * EE) / 256, 256, 0, stream>>>(Wv, up, o);
    outrow_kernel <<<(BB * EE) / 256, 256, 0, stream>>>(Wo, o, orow);
    bcast_kernel  <<<(BB * SS * EE / 4) / 256, 256, 0, stream>>>(orow, (float4*)d_out);
}